// GCN_47485158424802
// MI455X (gfx1250) — compile-verified
//
#include <hip/hip_runtime.h>

// ---------------- problem constants (match reference) ----------------
#define N_NODES 100000
#define N_EDGES 1600000
#define FDIM    64          // IN_DIM == HIDDEN == 64
#define OUT_DIM 2
#define KCHEB   4

typedef float v2f __attribute__((ext_vector_type(2)));
typedef float v8f __attribute__((ext_vector_type(8)));

// ---------------- utility kernels ----------------
__global__ void zero_f4(float4* __restrict__ p, int n4) {
  int i = blockIdx.x * blockDim.x + threadIdx.x;
  if (i < n4) p[i] = make_float4(0.f, 0.f, 0.f, 0.f);
}

__global__ void deg_kernel(const int* __restrict__ src, float* __restrict__ deg, int n_edges) {
  int i = blockIdx.x * blockDim.x + threadIdx.x;
  if (i < n_edges) atomicAdd(&deg[src[i]], 1.0f);
}

__global__ void dinv_kernel(const float* __restrict__ deg, float* __restrict__ dinv, int n) {
  int i = blockIdx.x * blockDim.x + threadIdx.x;
  if (i < n) {
    float d = deg[i];
    dinv[i] = (d > 0.f) ? rsqrtf(fmaxf(d, 1e-12f)) : 0.f;
  }
}

// Sparse propagation: acc[dst] += (-dinv[src]*dinv[dst]) * h[src]
// One wave32 per edge; lane l carries features {2l, 2l+1} as a float2.
// Gather: 256B contiguous per wave. Scatter: 64 global_atomic_add_f32 per edge,
// 256B contiguous per wave. This is the bandwidth-dominant phase (~5GB total).
__global__ void prop_kernel(const float* __restrict__ h,
                            const int*   __restrict__ src,
                            const int*   __restrict__ dst,
                            const float* __restrict__ dinv,
                            float*       __restrict__ acc,
                            int n_edges) {
  int gid  = blockIdx.x * blockDim.x + threadIdx.x;
  int e    = gid >> 5;
  int lane = threadIdx.x & 31;
  if (e >= n_edges) return;
  int s = src[e];
  int d = dst[e];
  float coeff = -dinv[s] * dinv[d];
  float2 v = ((const float2*)(h + (size_t)s * FDIM))[lane];
  float* a = acc + (size_t)d * FDIM + 2 * lane;
  atomicAdd(a + 0, coeff * v.x);
  atomicAdd(a + 1, coeff * v.y);
}

// Chebyshev recurrence tail: T = 2*T - prev   (T currently holds prop result)
__global__ void combine_kernel(float4* __restrict__ T, const float4* __restrict__ prev, int n4) {
  int i = blockIdx.x * blockDim.x + threadIdx.x;
  if (i < n4) {
    float4 t = T[i];
    float4 p = prev[i];
    t.x = 2.f * t.x - p.x;
    t.y = 2.f * t.y - p.y;
    t.z = 2.f * t.z - p.z;
    t.w = 2.f * t.w - p.w;
    T[i] = t;
  }
}

// Fused dense reduction: out = opt_relu( sum_{k=0..3} A_k @ W[k] )
//   A_k : [n_nodes x 64] fp32,  W : [4 x 64 x F_OUT] fp32,  out : [n_nodes x F_OUT]
// One wave per 16-row stripe covering ALL F_OUT columns (NT 16x16 tiles):
// A fragments are loaded once and reused for every N-tile.
// V_WMMA_F32_16X16X4_F32, K=64 in steps of 4, x4 Chebyshev terms.
// F_OUT and RELU are compile-time so all addressing folds to base+imm offsets
// and the partial-tile mask (F_OUT=2) is a clamp+cndmask, never an EXEC branch.
//
// Fragment layouts (ISA 7.12.2, 32-bit):
//   A 16x4 : lanes 0-15 M=lane, lanes 16-31 M=lane-16; VGPR v holds K = 2*half + v
//   B 4x16 : lanes hold N=lane&15; VGPR v holds K = 2*half + v (mirror of A)
//   C/D    : VGPR r -> M = r + 8*half, N = lane&15
template <int F_OUT, bool RELU>
__global__ __launch_bounds__(256) void cheb_gemm_kernel(const float* __restrict__ A0,
                                                        const float* __restrict__ A1,
                                                        const float* __restrict__ A2,
                                                        const float* __restrict__ A3,
                                                        const float* __restrict__ W,
                                                        float* __restrict__ out,
                                                        int n_nodes) {
  constexpr int NT = (F_OUT + 15) / 16;                // N-tiles per stripe
  int wave = (blockIdx.x * blockDim.x + threadIdx.x) >> 5;
  int lane = threadIdx.x & 31;
  int mtiles = n_nodes >> 4;                           // 100000 / 16 = 6250
  if (wave >= mtiles) return;                          // wave-uniform: EXEC stays all-ones

  int mbase = wave << 4;
  int half  = lane >> 4;
  int l16   = lane & 15;
  int arow  = mbase + l16;                             // A-fragment row for this lane

  const float* __restrict__ Ab[KCHEB] = {
      A0 + arow * FDIM, A1 + arow * FDIM, A2 + arow * FDIM, A3 + arow * FDIM};

  v8f c[NT];
#pragma unroll
  for (int nt = 0; nt < NT; ++nt) c[nt] = (v8f){};

#pragma unroll
  for (int k = 0; k < KCHEB; ++k) {
    const float* __restrict__ Ak = Ab[k];
    const float* __restrict__ Wk = W + k * FDIM * F_OUT;
#pragma unroll
    for (int kk = 0; kk < FDIM; kk += 4) {
      int ka = kk + 2 * half;                          // this lane's K for VGPR0
      v2f a;
      a.x = Ak[ka + 0];                                // -> global_load_b64
      a.y = Ak[ka + 1];
#pragma unroll
      for (int nt = 0; nt < NT; ++nt) {
        int bcol = nt * 16 + l16;
        v2f b;
        if constexpr ((F_OUT & 15) == 0) {
          b.x = Wk[(ka + 0) * F_OUT + bcol];
          b.y = Wk[(ka + 1) * F_OUT + bcol];
        } else {
          int bc   = (bcol < F_OUT) ? bcol : (F_OUT - 1);  // clamp: load always valid
          float bx = Wk[(ka + 0) * F_OUT + bc];
          float by = Wk[(ka + 1) * F_OUT + bc];
          b.x = (bcol < F_OUT) ? bx : 0.f;                 // v_cndmask, no EXEC change
          b.y = (bcol < F_OUT) ? by : 0.f;
        }
        c[nt] = __builtin_amdgcn_wmma_f32_16x16x4_f32(
                    /*neg_a=*/false, a, /*neg_b=*/false, b,
                    /*c_mod=*/(short)0, c[nt], /*reuse_a=*/false, /*reuse_b=*/false);
      }
    }
  }

#pragma unroll
  for (int nt = 0; nt < NT; ++nt) {
#pragma unroll
    for (int r = 0; r < 8; ++r) {
      int row = mbase + r + 8 * half;
      int col = nt * 16 + l16;
      if ((F_OUT & 15) == 0 || col < F_OUT) {
        float v = c[nt][r];
        if (RELU) v = fmaxf(v, 0.f);
        out[row * F_OUT + col] = v;
      }
    }
  }
}

// ---------------- host-side launch ----------------
static inline size_t align_up(size_t v, size_t a) { return (v + a - 1) & ~(a - 1); }

extern "C" void kernel_launch(void* const* d_in, const int* in_sizes, int n_in,
                              void* d_out, int out_size, void* d_ws, size_t ws_size,
                              hipStream_t stream) {
  (void)in_sizes; (void)n_in; (void)out_size; (void)ws_size;

  const float* x   = (const float*)d_in[0];            // [N,64]
  const int*   ei  = (const int*)d_in[1];              // [2,E]
  const float* W1  = (const float*)d_in[2];            // [4,64,64]
  const float* W2  = (const float*)d_in[3];            // [4,64,2]
  float*       out = (float*)d_out;                    // [N,2]

  const int* src = ei;                                 // edge_index[0]
  const int* dst = ei + N_EDGES;                       // edge_index[1]

  // workspace carve-up
  char* ws = (char*)d_ws;
  size_t off = 0;
  float* deg  = (float*)(ws + off); off = align_up(off + (size_t)N_NODES * 4, 4096);
  float* dinv = (float*)(ws + off); off = align_up(off + (size_t)N_NODES * 4, 4096);
  const size_t featBytes = (size_t)N_NODES * FDIM * 4;  // 25.6 MB
  float* T1 = (float*)(ws + off); off = align_up(off + featBytes, 4096);
  float* T2 = (float*)(ws + off); off = align_up(off + featBytes, 4096);
  float* T3 = (float*)(ws + off); off = align_up(off + featBytes, 4096);
  float* h  = (float*)(ws + off); off = align_up(off + featBytes, 4096);

  const int BLK = 256;
  const int nodef4   = N_NODES / 4;                    // 25000 float4 (deg/dinv zero)
  const int featf4   = N_NODES * (FDIM / 4);           // 1.6M float4
  const int degBlks  = (N_EDGES + BLK - 1) / BLK;
  const int nodeBlks = (N_NODES + BLK - 1) / BLK;
  const int propBlks = (int)(((size_t)N_EDGES * 32 + BLK - 1) / BLK);
  const int featBlks = (featf4 + BLK - 1) / BLK;
  const int zdegBlks = (nodef4 + BLK - 1) / BLK;
  const int mtiles   = N_NODES / 16;                   // 6250 waves per gemm
  const int gemmBlks = (mtiles * 32 + BLK - 1) / BLK;

  // --- degree + normalization (shared by both layers) ---
  zero_f4<<<zdegBlks, BLK, 0, stream>>>((float4*)deg, nodef4);
  deg_kernel<<<degBlks, BLK, 0, stream>>>(src, deg, N_EDGES);
  dinv_kernel<<<nodeBlks, BLK, 0, stream>>>(deg, dinv, N_NODES);

  // --- Layer 1: tx0 = x, tx1 = T1, tx2 = T2, tx3 = T3 ---
  zero_f4<<<featBlks, BLK, 0, stream>>>((float4*)T1, featf4);
  prop_kernel<<<propBlks, BLK, 0, stream>>>(x, src, dst, dinv, T1, N_EDGES);

  zero_f4<<<featBlks, BLK, 0, stream>>>((float4*)T2, featf4);
  prop_kernel<<<propBlks, BLK, 0, stream>>>(T1, src, dst, dinv, T2, N_EDGES);
  combine_kernel<<<featBlks, BLK, 0, stream>>>((float4*)T2, (const float4*)x, featf4);

  zero_f4<<<featBlks, BLK, 0, stream>>>((float4*)T3, featf4);
  prop_kernel<<<propBlks, BLK, 0, stream>>>(T2, src, dst, dinv, T3, N_EDGES);
  combine_kernel<<<featBlks, BLK, 0, stream>>>((float4*)T3, (const float4*)T1, featf4);

  cheb_gemm_kernel<FDIM, true><<<gemmBlks, BLK, 0, stream>>>(x, T1, T2, T3, W1, h, N_NODES);

  // --- Layer 2: tx0 = h (reuse T1..T3) ---
  zero_f4<<<featBlks, BLK, 0, stream>>>((float4*)T1, featf4);
  prop_kernel<<<propBlks, BLK, 0, stream>>>(h, src, dst, dinv, T1, N_EDGES);

  zero_f4<<<featBlks, BLK, 0, stream>>>((float4*)T2, featf4);
  prop_kernel<<<propBlks, BLK, 0, stream>>>(T1, src, dst, dinv, T2, N_EDGES);
  combine_kernel<<<featBlks, BLK, 0, stream>>>((float4*)T2, (const float4*)h, featf4);

  zero_f4<<<featBlks, BLK, 0, stream>>>((float4*)T3, featf4);
  prop_kernel<<<propBlks, BLK, 0, stream>>>(T2, src, dst, dinv, T3, N_EDGES);
  combine_kernel<<<featBlks, BLK, 0, stream>>>((float4*)T3, (const float4*)T1, featf4);

  cheb_gemm_kernel<OUT_DIM, false><<<gemmBlks, BLK, 0, stream>>>(h, T1, T2, T3, W2, out, N_NODES);
}